// FlipModelQubo_47141561041154
// MI455X (gfx1250) — compile-verified
//
#include <hip/hip_runtime.h>
#include <math.h>

// FlipModelQubo — N=4096, f32.
// term = sum_ij Q[i,j]*( m[i,j]*w_i*w_j + p_i*b_j + b_i*p_j + p_i*p_j )
//        + sum_i Q[i,i]*(p_i - p_i^2)  + 0.01*e/N * sum_i (p_i+eps)*log(1/(p_i+eps))
// where p=(1+cos a)/2, w=1-2p, b=diag(x)*w.
// Memory-bound: 128MB stream -> ~5.7us floor @ 23.3 TB/s.

typedef __attribute__((ext_vector_type(2))) float v2f;
typedef __attribute__((ext_vector_type(4))) float v4f;
typedef __attribute__((ext_vector_type(8))) float v8f;

#define WS_ACC_DIAG 1
#define WS_ACC_ENT  2
#define WS_VEC_OFF  16

// Wave32 reduction via V_WMMA_F32_16X16X4_F32:
// A(16x4): v0 = per-lane partial (lanes 0-15 -> K=0, lanes 16-31 -> K=2), v1 = 0.
// B(4x16) = ones  =>  D[m,n] = acc[m] + acc[m+16]  (independent of n).
// Per-lane: sum 8 D VGPRs = sum over 8 rows; xor-16 partner holds the other 8.
__device__ __forceinline__ float waveReduceWMMA(float v) {
  v2f a; a.x = v;    a.y = 0.0f;
  v2f b; b.x = 1.0f; b.y = 1.0f;
  v8f c = {};
  c = __builtin_amdgcn_wmma_f32_16x16x4_f32(
      /*neg_a=*/false, a, /*neg_b=*/false, b,
      /*c_mod=*/(short)0, c, /*reuse_a=*/false, /*reuse_b=*/false);
  float s = ((c[0] + c[1]) + (c[2] + c[3])) + ((c[4] + c[5]) + (c[6] + c[7]));
  s += __shfl_xor(s, 16, 32);
  return s;  // full 32-lane sum, valid in every lane
}

// Block reduction; result valid on threadIdx.x == 0. Safe to call repeatedly.
__device__ __forceinline__ float blockReduce(float v) {
  __shared__ float sm[32];
  float w = waveReduceWMMA(v);
  int lane = threadIdx.x & 31;
  int wid  = threadIdx.x >> 5;
  if (lane == 0) sm[wid] = w;
  __syncthreads();
  float total = 0.0f;
  if (threadIdx.x == 0) {
    int nw = (blockDim.x + 31) >> 5;
    for (int k = 0; k < nw; ++k) total += sm[k];
  }
  __syncthreads();  // protect sm against a subsequent call
  return total;
}

__global__ void qubo_init(float* __restrict__ ws) {
  if (threadIdx.x < 8) ws[threadIdx.x] = 0.0f;
}

// Per-index vectors + diagonal/entropy scalars (O(N)).
__global__ void qubo_vec(const float* __restrict__ x, const float* __restrict__ Q,
                         const float* __restrict__ alphas, float* __restrict__ ws, int N) {
  float* pv = ws + WS_VEC_OFF;
  float* wv = pv + N;
  float* bv = wv + N;
  int i = blockIdx.x * blockDim.x + threadIdx.x;
  float dterm = 0.0f, eterm = 0.0f;
  if (i < N) {
    float p = 0.5f * (1.0f + cosf(alphas[i]));
    float w = 1.0f - 2.0f * p;
    float d = x[(size_t)i * N + i];
    pv[i] = p;
    wv[i] = w;
    bv[i] = d * w;
    dterm = Q[(size_t)i * N + i] * (p - p * p);
    float pe = p + 1e-14f;
    eterm = pe * logf(1.0f / pe);
  }
  float dsum = blockReduce(dterm);
  float esum = blockReduce(eterm);
  if (threadIdx.x == 0) {
    atomicAdd(ws + WS_ACC_DIAG, dsum);
    atomicAdd(ws + WS_ACC_ENT,  esum);
  }
}

// Main streaming pass: one float4 column-group per thread, one row per blockIdx.y.
// Q/x streamed non-temporally (read-once); p/w/b vectors stay hot in cache.
__global__ void qubo_main(const float* __restrict__ x, const float* __restrict__ Q,
                          const float* __restrict__ ws, float* __restrict__ partials, int N) {
  const float* pv = ws + WS_VEC_OFF;
  const float* wv = pv + N;
  const float* bv = wv + N;
  const int ncol = N >> 2;
  const int i  = blockIdx.y;
  const int jg = blockIdx.x * blockDim.x + threadIdx.x;
  float acc = 0.0f;
  if (jg < ncol) {
    size_t off = (size_t)i * (size_t)N + (size_t)jg * 4;
    v4f q  = __builtin_nontemporal_load((const v4f*)(Q + off));
    v4f m  = __builtin_nontemporal_load((const v4f*)(x + off));
    v4f pj = *(const v4f*)(pv + (size_t)jg * 4);
    v4f wj = *(const v4f*)(wv + (size_t)jg * 4);
    v4f bj = *(const v4f*)(bv + (size_t)jg * 4);
    float pi = pv[i], wi = wv[i], bi = bv[i];
    v4f t = q * (m * (wi * wj) + pi * bj + bi * pj + pi * pj);
    acc = (t.x + t.y) + (t.z + t.w);
  }
  float bs = blockReduce(acc);
  if (threadIdx.x == 0)
    partials[(size_t)blockIdx.y * gridDim.x + blockIdx.x] = bs;
}

// Reduce per-block partials + fold in diagonal & entropy terms.
__global__ void qubo_finalize(const float* __restrict__ ws, const float* __restrict__ partials,
                              int nparts, float* __restrict__ out, int N) {
  float acc = 0.0f;
  for (int k = threadIdx.x; k < nparts; k += blockDim.x) acc += partials[k];
  float total = blockReduce(acc);
  if (threadIdx.x == 0) {
    // norm = N * ln(e)/e = N/e  ->  E_p = 0.01 * ent_sum * e / N
    float Ep = 0.01f * ws[WS_ACC_ENT] * 2.71828182845904523f / (float)N;
    out[0] = total + ws[WS_ACC_DIAG] + Ep;
  }
}

extern "C" void kernel_launch(void* const* d_in, const int* in_sizes, int n_in,
                              void* d_out, int out_size, void* d_ws, size_t ws_size,
                              hipStream_t stream) {
  const float* x      = (const float*)d_in[0];   // (1,N,N)
  const float* Q      = (const float*)d_in[1];   // (N,N)
  const float* alphas = (const float*)d_in[2];   // (1,N)
  float* out = (float*)d_out;
  float* ws  = (float*)d_ws;

  const int N    = in_sizes[2];
  const int BT   = 256;
  const int ncol = N >> 2;
  const int gx   = (ncol + BT - 1) / BT;
  const int nparts = gx * N;
  float* partials = ws + WS_VEC_OFF + 3 * (size_t)N;

  qubo_init<<<1, 32, 0, stream>>>(ws);
  qubo_vec<<<(N + BT - 1) / BT, BT, 0, stream>>>(x, Q, alphas, ws, N);
  dim3 grid(gx, N, 1);
  qubo_main<<<grid, BT, 0, stream>>>(x, Q, ws, partials, N);
  qubo_finalize<<<1, BT, 0, stream>>>(ws, partials, nparts, out, N);
}